// GATv2Layer_83004537962839
// MI455X (gfx1250) — compile-verified
//
#include <hip/hip_runtime.h>
#include <hip/hip_bf16.h>

#define NN      10000
#define EE      320000
#define EF      (EE + NN)        // edges + self loops
#define IN_DIM  256
#define HH      4
#define CC      64
#define HC      256              // HH * CC
#define ED      64
#define EPS     1e-5f
#define NEG_SLOPE 0.2f

typedef __attribute__((ext_vector_type(2))) float v2f;
typedef __attribute__((ext_vector_type(8))) float v8f;

static __device__ __forceinline__ v8f wmma_f32x4(v2f a, v2f b, v8f c) {
    return __builtin_amdgcn_wmma_f32_16x16x4_f32(
        /*neg_a=*/false, a, /*neg_b=*/false, b,
        /*c_mod=*/(short)0, c, /*reuse_a=*/false, /*reuse_b=*/false);
}

// ---------------------------------------------------------------- utilities
__global__ void fill_kernel(float* __restrict__ p, float v, int n) {
    int i = blockIdx.x * blockDim.x + threadIdx.x;
    if (i < n) p[i] = v;
}

static __device__ __forceinline__ void atomicMaxF(float* addr, float val) {
    unsigned int* ai = (unsigned int*)addr;
    unsigned int old = *ai;
    while (true) {
        float f = __uint_as_float(old);
        if (f >= val) break;
        unsigned int assumed = old;
        old = atomicCAS(ai, assumed, __float_as_uint(val));
        if (old == assumed) break;
    }
}

// LDS-tiled transpose: in[R][C] -> out[C][R].  R, C multiples of 16.
// grid = (C/16, R/16), block = (16,16)
__global__ void transpose_kernel(const float* __restrict__ in,
                                 float* __restrict__ out, int R, int C) {
    __shared__ float tile[16][17];
    int tx = threadIdx.x, ty = threadIdx.y;
    int r = blockIdx.y * 16 + ty;
    int c = blockIdx.x * 16 + tx;
    tile[ty][tx] = in[(size_t)r * C + c];
    __syncthreads();
    int ro = blockIdx.x * 16 + ty;   // output row  (= input col)
    int co = blockIdx.y * 16 + tx;   // output col  (= input row)
    out[(size_t)ro * R + co] = tile[tx][ty];
}

// ------------------------------------------- self-loop attr mean (fill_value='mean')
__global__ void loop_accum_kernel(const float* __restrict__ edge_attr,
                                  const int* __restrict__ dst,
                                  float* __restrict__ loopsum,   // [N*ED]
                                  float* __restrict__ cnt) {     // [N]
    int idx = blockIdx.x * blockDim.x + threadIdx.x;
    if (idx >= EE * ED) return;
    int e = idx >> 6;           // / ED
    int d = idx & (ED - 1);
    int t = dst[e];
    atomicAdd(&loopsum[t * ED + d], edge_attr[idx]);
    if (d == 0) atomicAdd(&cnt[t], 1.0f);
}

__global__ void loop_norm_kernel(float* __restrict__ loopattr,   // in-place /= cnt
                                 const float* __restrict__ cnt) {
    int idx = blockIdx.x * blockDim.x + threadIdx.x;
    if (idx >= NN * ED) return;
    loopattr[idx] = loopattr[idx] / fmaxf(cnt[idx >> 6], 1.0f);
}

// ------------------------------------------- node projections: xl = x@Wl.T+bl, xr = x@Wr.T+br
// K-major operands: xT[IN][NN], Wt[IN][HC].  One wave = one 16x16 tile, K=256.
// Every A/B load is 64B-contiguous per half-wave.
__global__ void node_proj_kernel(const float* __restrict__ xT,
                                 const float* __restrict__ Wlt, const float* __restrict__ bl,
                                 const float* __restrict__ Wrt, const float* __restrict__ br,
                                 float* __restrict__ xl, float* __restrict__ xr) {
    const int WAVES_PER_BLK = 4;
    int wid = blockIdx.x * WAVES_PER_BLK + (threadIdx.x >> 5);
    const int TOTAL = (NN / 16) * (HC / 16) * 2;   // 20000 waves
    if (wid >= TOTAL) return;

    int sel = wid & 1;           // 0 -> Wl/xl, 1 -> Wr/xr
    int rem = wid >> 1;
    int jt  = rem & 15;          // column tile (HC/16 = 16)
    int it  = rem >> 4;          // row tile

    int lane = threadIdx.x & 31;
    int half = lane >> 4;
    int ln   = lane & 15;

    const float* Wt = sel ? Wrt : Wlt;
    const float* bb = sel ? br : bl;
    float* outp     = sel ? xr : xl;

    int rowm = it * 16 + ln;
    int col  = jt * 16 + ln;

    v8f acc = {};
    for (int k0 = 0; k0 < IN_DIM; k0 += 4) {
        int ka = k0 + 2 * half;                 // VGPR0: K=2*half, VGPR1: K=2*half+1
        v2f a; a[0] = xT[(size_t)ka * NN + rowm];
               a[1] = xT[(size_t)(ka + 1) * NN + rowm];
        v2f b; b[0] = Wt[(size_t)ka * HC + col];
               b[1] = Wt[(size_t)(ka + 1) * HC + col];
        acc = wmma_f32x4(a, b, acc);
    }

    float bv = bb[col];
    for (int r = 0; r < 8; ++r) {
        int row = it * 16 + half * 8 + r;       // D: lanes 0-15 -> M=r, 16-31 -> M=r+8
        outp[(size_t)row * HC + col] = acc[r] + bv;
    }
}

// ------------------------------------------- fused edge kernel:
// per wave: 16-edge tile; ep via WMMA (K=64) with A hoisted into 32 regs and
// K-major B (Wet[ED][HC]); + xl[src] + xr[dst], leaky-relu, dot with att,
// half-wave shuffle reduce -> logits [EF, H].  ep never touches HBM.
__global__ void edge_logits_kernel(const float* __restrict__ edge_attr,
                                   const float* __restrict__ loopattr,
                                   const int* __restrict__ src,
                                   const int* __restrict__ dst,
                                   const float* __restrict__ Wet,  // [ED][HC] K-major
                                   const float* __restrict__ att,  // [HC] flat
                                   const float* __restrict__ xl,
                                   const float* __restrict__ xr,
                                   float* __restrict__ logits) {   // [EF, H]
    const int WAVES_PER_BLK = 4;
    int t = blockIdx.x * WAVES_PER_BLK + (threadIdx.x >> 5);
    const int TILES = EF / 16;                  // 20625
    if (t >= TILES) return;

    int lane = threadIdx.x & 31;
    int half = lane >> 4;
    int ln   = lane & 15;

    int arow = t * 16 + ln;
    const float* aptr = (arow < EE) ? (edge_attr + (size_t)arow * ED)
                                    : (loopattr + (size_t)(arow - EE) * ED);

    // hoist this lane's A fragment (32 of the 64 row values) into registers
    float areg[32];
    #pragma unroll
    for (int k0 = 0; k0 < ED; k0 += 4) {
        int ka = k0 + 2 * half;
        areg[(k0 >> 1) + 0] = aptr[ka];
        areg[(k0 >> 1) + 1] = aptr[ka + 1];
    }

    int dr[8], srcs[8], dsts[8];
    for (int r = 0; r < 8; ++r) {
        int row = t * 16 + half * 8 + r;
        dr[r] = row;
        if (row < EE) { srcs[r] = src[row]; dsts[r] = dst[row]; }
        else          { srcs[r] = row - EE; dsts[r] = row - EE; }
    }
    // prefetch the gathered projection rows (global_prefetch_b8)
    #pragma unroll
    for (int r = 0; r < 8; ++r) {
        __builtin_prefetch(&xl[(size_t)srcs[r] * HC], 0, 1);
        __builtin_prefetch(&xr[(size_t)dsts[r] * HC], 0, 1);
    }

    float logacc[8] = {0.f, 0.f, 0.f, 0.f, 0.f, 0.f, 0.f, 0.f};

    for (int j = 0; j < HC / 16; ++j) {         // 16 column tiles
        int col = j * 16 + ln;
        v8f acc = {};
        #pragma unroll
        for (int k0 = 0; k0 < ED; k0 += 4) {
            int ka = k0 + 2 * half;
            v2f a; a[0] = areg[(k0 >> 1) + 0];
                   a[1] = areg[(k0 >> 1) + 1];
            v2f b; b[0] = Wet[(size_t)ka * HC + col];
                   b[1] = Wet[(size_t)(ka + 1) * HC + col];
            acc = wmma_f32x4(a, b, acc);
        }
        float attv = att[col];
        for (int r = 0; r < 8; ++r) {
            float mv = acc[r] + xl[(size_t)srcs[r] * HC + col]
                              + xr[(size_t)dsts[r] * HC + col];
            mv = (mv >= 0.f) ? mv : NEG_SLOPE * mv;
            logacc[r] += mv * attv;
        }
        if ((j & 3) == 3) {                     // head complete: 64 cols consumed
            int h = j >> 2;
            for (int r = 0; r < 8; ++r) {
                float v = logacc[r];
                v += __shfl_xor(v, 8, 16);
                v += __shfl_xor(v, 4, 16);
                v += __shfl_xor(v, 2, 16);
                v += __shfl_xor(v, 1, 16);
                if (ln == 0) logits[(size_t)dr[r] * HH + h] = v;
                logacc[r] = 0.f;
            }
        }
    }
}

// ------------------------------------------- segment softmax pieces
static __device__ __forceinline__ int dst_of(int e, const int* dst) {
    return (e < EE) ? dst[e] : (e - EE);
}

__global__ void seg_max_kernel(const float* __restrict__ logits,
                               const int* __restrict__ dst,
                               float* __restrict__ mx) {           // [N*H]
    int idx = blockIdx.x * blockDim.x + threadIdx.x;
    if (idx >= EF * HH) return;
    int e = idx >> 2, h = idx & 3;
    atomicMaxF(&mx[dst_of(e, dst) * HH + h], logits[idx]);
}

__global__ void seg_exp_sum_kernel(float* __restrict__ logits,     // -> exp in place
                                   const int* __restrict__ dst,
                                   const float* __restrict__ mx,
                                   float* __restrict__ den) {      // [N*H]
    int idx = blockIdx.x * blockDim.x + threadIdx.x;
    if (idx >= EF * HH) return;
    int e = idx >> 2, h = idx & 3;
    float ex = __expf(logits[idx] - mx[dst_of(e, dst) * HH + h]);
    logits[idx] = ex;
    atomicAdd(&den[dst_of(e, dst) * HH + h], ex);
}

__global__ void alpha_kernel(const float* __restrict__ exlog,
                             const int* __restrict__ dst,
                             const float* __restrict__ den,
                             float* __restrict__ alpha) {          // [EF*H] -> d_out tail
    int idx = blockIdx.x * blockDim.x + threadIdx.x;
    if (idx >= EF * HH) return;
    int e = idx >> 2, h = idx & 3;
    alpha[idx] = exlog[idx] / den[dst_of(e, dst) * HH + h];
}

// ------------------------------------------- out[dst] += xl[src] * alpha
__global__ void aggregate_kernel(const float* __restrict__ xl,
                                 const float* __restrict__ alpha,
                                 const int* __restrict__ src,
                                 const int* __restrict__ dst,
                                 float* __restrict__ outacc) {     // [N*HC]
    int idx = blockIdx.x * blockDim.x + threadIdx.x;
    if (idx >= EF * HC) return;
    int e = idx >> 8;          // / HC
    int c = idx & (HC - 1);
    int h = c >> 6;            // / CC
    int s = (e < EE) ? src[e] : (e - EE);
    int d = dst_of(e, dst);
    float a = alpha[(size_t)e * HH + h];
    atomicAdd(&outacc[(size_t)d * HC + c], xl[(size_t)s * HC + c] * a);
}

// ------------------------------------------- bias + SiLU + LayerNorm
__global__ void finalize_kernel(const float* __restrict__ outacc,
                                const float* __restrict__ bias,
                                const float* __restrict__ gamma,
                                const float* __restrict__ beta,
                                float* __restrict__ out) {         // [N*HC]
    int n = blockIdx.x;
    int c = threadIdx.x;       // blockDim.x == HC == 256
    float h = outacc[(size_t)n * HC + c] + bias[c];
    h = h / (1.0f + __expf(-h));          // SiLU = h * sigmoid(h)

    float s = h, s2 = h * h;
    for (int off = 16; off >= 1; off >>= 1) {
        s  += __shfl_xor(s,  off, 32);
        s2 += __shfl_xor(s2, off, 32);
    }
    __shared__ float ssum[8], ssq[8];
    int wid = threadIdx.x >> 5, lnn = threadIdx.x & 31;
    if (lnn == 0) { ssum[wid] = s; ssq[wid] = s2; }
    __syncthreads();
    if (threadIdx.x == 0) {
        float a = 0.f, b = 0.f;
        for (int i = 0; i < 8; ++i) { a += ssum[i]; b += ssq[i]; }
        ssum[0] = a; ssq[0] = b;
    }
    __syncthreads();
    float mu  = ssum[0] * (1.0f / HC);
    float var = ssq[0] * (1.0f / HC) - mu * mu;
    out[(size_t)n * HC + c] = (h - mu) * rsqrtf(var + EPS) * gamma[c] + beta[c];
}

// ---------------------------------------------------------------- launcher
static inline int cdiv(long long a, long long b) { return (int)((a + b - 1) / b); }

extern "C" void kernel_launch(void* const* d_in, const int* in_sizes, int n_in,
                              void* d_out, int out_size, void* d_ws, size_t ws_size,
                              hipStream_t stream) {
    const float* x     = (const float*)d_in[0];
    const int*   ei    = (const int*)  d_in[1];
    const float* eattr = (const float*)d_in[2];
    const float* Wl    = (const float*)d_in[3];
    const float* bl    = (const float*)d_in[4];
    const float* Wr    = (const float*)d_in[5];
    const float* br    = (const float*)d_in[6];
    const float* We    = (const float*)d_in[7];
    const float* att   = (const float*)d_in[8];
    const float* bias  = (const float*)d_in[9];
    const float* gamma = (const float*)d_in[10];
    const float* beta  = (const float*)d_in[11];

    const int* src = ei;
    const int* dst = ei + EE;

    // workspace layout (floats): ~50 MB total
    float* ws       = (float*)d_ws;
    float* cnt      = ws;                           // N
    float* loopattr = cnt + NN;                     // N*ED   (sums, then normalized)
    float* xl       = loopattr + (size_t)NN * ED;   // N*HC
    float* xr       = xl + (size_t)NN * HC;         // N*HC
    float* logits   = xr + (size_t)NN * HC;         // EF*H   (-> exp in place)
    float* mx       = logits + (size_t)EF * HH;     // N*H
    float* den      = mx + (size_t)NN * HH;         // N*H
    float* outacc   = den + (size_t)NN * HH;        // N*HC
    float* xT       = outacc + (size_t)NN * HC;     // IN*NN   (x transposed, K-major)
    float* Wlt      = xT + (size_t)IN_DIM * NN;     // IN*HC
    float* Wrt      = Wlt + (size_t)IN_DIM * HC;    // IN*HC
    float* Wet      = Wrt + (size_t)IN_DIM * HC;    // ED*HC

    float* outh  = (float*)d_out;                   // [N*HC]
    float* alpha = outh + (size_t)NN * HC;          // [EF*H]

    const int T = 256;

    // init: zero {cnt, loopattr} (contiguous), {den, outacc} (contiguous), mx = -inf
    fill_kernel<<<cdiv(NN + NN * ED, T), T, 0, stream>>>(cnt, 0.0f, NN + NN * ED);
    fill_kernel<<<cdiv(NN * HH + NN * HC, T), T, 0, stream>>>(den, 0.0f, NN * HH + NN * HC);
    fill_kernel<<<cdiv(NN * HH, T), T, 0, stream>>>(mx, -3.0e38f, NN * HH);

    // K-major staging (one-time transposes; all dims multiples of 16)
    {
        dim3 blk(16, 16);
        transpose_kernel<<<dim3(IN_DIM / 16, NN / 16), blk, 0, stream>>>(x,  xT,  NN,     IN_DIM);
        transpose_kernel<<<dim3(IN_DIM / 16, HC / 16), blk, 0, stream>>>(Wl, Wlt, HC,     IN_DIM);
        transpose_kernel<<<dim3(IN_DIM / 16, HC / 16), blk, 0, stream>>>(Wr, Wrt, HC,     IN_DIM);
        transpose_kernel<<<dim3(ED / 16,     HC / 16), blk, 0, stream>>>(We, Wet, HC,     ED);
    }

    // self-loop attrs = mean of incoming edge attrs
    loop_accum_kernel<<<cdiv((long long)EE * ED, T), T, 0, stream>>>(eattr, dst, loopattr, cnt);
    loop_norm_kernel<<<cdiv(NN * ED, T), T, 0, stream>>>(loopattr, cnt);

    // node projections (WMMA): 20000 waves, 4 waves/block
    node_proj_kernel<<<cdiv((NN / 16) * (HC / 16) * 2, 4), 128, 0, stream>>>(
        xT, Wlt, bl, Wrt, br, xl, xr);

    // fused edge projection + attention logits (WMMA): 20625 waves
    edge_logits_kernel<<<cdiv(EF / 16, 4), 128, 0, stream>>>(
        eattr, loopattr, src, dst, Wet, att, xl, xr, logits);

    // segment softmax over dst
    seg_max_kernel<<<cdiv((long long)EF * HH, T), T, 0, stream>>>(logits, dst, mx);
    seg_exp_sum_kernel<<<cdiv((long long)EF * HH, T), T, 0, stream>>>(logits, dst, mx, den);
    alpha_kernel<<<cdiv((long long)EF * HH, T), T, 0, stream>>>(logits, dst, den, alpha);

    // weighted aggregation
    aggregate_kernel<<<cdiv((long long)EF * HC, T), T, 0, stream>>>(xl, alpha, src, dst, outacc);

    // bias + SiLU + LayerNorm
    finalize_kernel<<<NN, HC, 0, stream>>>(outacc, bias, gamma, beta, outh);
}